// GCSS2D_33758442947268
// MI455X (gfx1250) — compile-verified
//
#include <hip/hip_runtime.h>
#include <math.h>

// Problem constants (match reference)
#define C_INx   96
#define D_INN   192      // EXPAND * C_IN
#define D_ST    16
#define DT_R    6
#define KDIR    4
#define NB      4        // batch
#define HHx     64
#define WWx     64
#define LLx     4096     // H*W
#define NROW    38       // DT_RANK + 2*D_STATE

typedef float v2f __attribute__((ext_vector_type(2)));
typedef float v8f __attribute__((ext_vector_type(8)));

__device__ __forceinline__ float silu_f(float x) { return x / (1.f + __expf(-x)); }
__device__ __forceinline__ float softplus_f(float x) {
  return (x > 20.f) ? x : log1pf(__expf(x));
}
// Scan-direction pixel mapping: l (scan index) -> p (row-major pixel index)
__device__ __forceinline__ int dir_map(int k, int l) {
  switch (k) {
    case 0: return l;                                   // row-major
    case 1: return ((l & 63) << 6) | (l >> 6);          // col-major (W,H transpose)
    case 2: return (LLx - 1) - l;                       // reversed row-major
    default: { int t = (LLx - 1) - l; return ((t & 63) << 6) | (t >> 6); }
  }
}

// ---------------------------------------------------------------------------
// Kernel 1: xz = xp @ W_in.T   (M=16384, N=384, K=96)  fp32 WMMA 16x16x4
// Splits result into x_in (B,192,H,W) and z (B*L,192).
// ---------------------------------------------------------------------------
__global__ __launch_bounds__(128) void k_gemm_in(const float* __restrict__ x,
                                                 const float* __restrict__ W_in,
                                                 float* __restrict__ x_in,
                                                 float* __restrict__ z) {
  const int lane = threadIdx.x & 31;
  const int tile = blockIdx.x * 4 + (threadIdx.x >> 5);
  const int NT = (2 * D_INN) / 16;            // 24 column tiles
  const int mt = tile / NT, nt = tile % NT;
  const int m0 = mt * 16, n0 = nt * 16;
  const int half = lane >> 4, mr = lane & 15;
  // A operand row / B operand column handled by this lane
  const int m = m0 + mr;
  const int b = m >> 12, l = m & 4095;
  const int ncol = n0 + mr;
  v8f acc = {};
  for (int kb = 0; kb < C_INx; kb += 4) {
    const int k0 = kb + 2 * half;            // lanes 16-31 hold K+2 (ISA A/B layout)
    v2f a, bv;
    a.x  = x[((size_t)b * C_INx + k0) * LLx + l];       // xp[m, k] = x[b,k,h,w]
    a.y  = x[((size_t)b * C_INx + k0 + 1) * LLx + l];
    bv.x = W_in[ncol * C_INx + k0];                     // W_in.T[k, n]
    bv.y = W_in[ncol * C_INx + k0 + 1];
    acc = __builtin_amdgcn_wmma_f32_16x16x4_f32(false, a, false, bv,
                                                (short)0, acc, false, false);
  }
  // D layout: VGPR r -> (M = m0+r+8*half, N = n0+(lane&15))
  const int col = n0 + mr;
  for (int r = 0; r < 8; ++r) {
    const int mrow = m0 + r + 8 * half;
    const int bb = mrow >> 12, ll = mrow & 4095;
    const float v = acc[r];
    if (col < D_INN)
      x_in[((size_t)bb * D_INN + col) * LLx + ll] = v;
    else
      z[(size_t)mrow * D_INN + (col - D_INN)] = v;
  }
}

// ---------------------------------------------------------------------------
// Kernel 2: depthwise 3x3 conv (SAME) + bias + SiLU
// ---------------------------------------------------------------------------
__global__ void k_conv(const float* __restrict__ x_in, const float* __restrict__ cw,
                       const float* __restrict__ cb, float* __restrict__ xc) {
  const int idx = blockIdx.x * blockDim.x + threadIdx.x;   // (b*192+ch)*4096 + l
  if (idx >= NB * D_INN * LLx) return;
  const int l = idx & 4095;
  const int bc = idx >> 12;
  const int ch = bc % D_INN;
  const int h = l >> 6, w = l & 63;
  float s = cb[ch];
  const float* base = x_in + (size_t)bc * LLx;
  for (int ky = 0; ky < 3; ++ky) {
    const int hy = h + ky - 1;
    if (hy < 0 || hy >= HHx) continue;
    for (int kx = 0; kx < 3; ++kx) {
      const int wx = w + kx - 1;
      if (wx < 0 || wx >= WWx) continue;
      s += base[hy * WWx + wx] * cw[ch * 9 + ky * 3 + kx];
    }
  }
  xc[idx] = silu_f(s);
}

// ---------------------------------------------------------------------------
// Kernel 3: x_dbl[b,k,c,l] = sum_d x_proj_w[k,c,d] * xs[b,k,d,l]
// xs never materialized: indexed directly out of xc via dir_map.
// ---------------------------------------------------------------------------
__global__ void k_xdbl(const float* __restrict__ xc, const float* __restrict__ xpw,
                       float* __restrict__ xdbl) {
  const int idx = blockIdx.x * blockDim.x + threadIdx.x;   // ((bk*38)+c)*4096 + l
  if (idx >= NB * KDIR * NROW * LLx) return;
  const int l = idx & 4095;
  const int t = idx >> 12;
  const int c = t % NROW;
  const int bk = t / NROW;
  const int k = bk & 3, b = bk >> 2;
  const int p = dir_map(k, l);
  const float* w = xpw + ((size_t)k * NROW + c) * D_INN;
  const float* src = xc + (size_t)b * D_INN * LLx + p;
  float s = 0.f;
  for (int d = 0; d < D_INN; ++d) s += w[d] * src[(size_t)d * LLx];
  xdbl[idx] = s;
}

// ---------------------------------------------------------------------------
// Kernel 4: selective scan. One lane per (b,k,d); h[16] lives in VGPRs.
// delta fused: softplus(dts . dt_w + dt_b). B/C/dts loads are wave-uniform.
// ---------------------------------------------------------------------------
__global__ __launch_bounds__(192) void k_scan(const float* __restrict__ xc,
                                              const float* __restrict__ xdbl,
                                              const float* __restrict__ dt_w,
                                              const float* __restrict__ dt_b,
                                              const float* __restrict__ A_logs,
                                              const float* __restrict__ Ds,
                                              float* __restrict__ out_y) {
  const int d = threadIdx.x;              // 0..191 (consecutive d within wave)
  const int bk = blockIdx.x;              // 0..15
  const int k = bk & 3, b = bk >> 2;
  const int kd = k * D_INN + d;
  float A[D_ST], h[D_ST];
  for (int n = 0; n < D_ST; ++n) {
    A[n] = -__expf(A_logs[kd * D_ST + n]);
    h[n] = 0.f;
  }
  float wdt[DT_R];
  for (int r = 0; r < DT_R; ++r) wdt[r] = dt_w[kd * DT_R + r];
  const float dtb = dt_b[kd];
  const float Dsv = Ds[kd];
  const float* u_base = xc + ((size_t)b * D_INN + d) * LLx;
  const float* xd = xdbl + (size_t)bk * NROW * LLx;
  float* out = out_y + ((size_t)bk * D_INN + d) * LLx;
  for (int l = 0; l < LLx; ++l) {
    const int p = dir_map(k, l);
    const float u = u_base[p];
    float dacc = dtb;
    for (int r = 0; r < DT_R; ++r) dacc += wdt[r] * xd[r * LLx + l];
    const float delta = softplus_f(dacc);
    const float du = delta * u;
    float y = 0.f;
    for (int n = 0; n < D_ST; ++n) {
      const float Bv = xd[(DT_R + n) * LLx + l];
      const float Cv = xd[(DT_R + D_ST + n) * LLx + l];
      h[n] = h[n] * __expf(delta * A[n]) + du * Bv;
      y += h[n] * Cv;
    }
    out[l] = y + Dsv * u;
  }
}

// ---------------------------------------------------------------------------
// Kernel 5: merge 4 directions (with index remap) + LayerNorm + SiLU gate
// ---------------------------------------------------------------------------
__global__ __launch_bounds__(192) void k_combine(const float* __restrict__ out_y,
                                                 const float* __restrict__ z,
                                                 const float* __restrict__ ln_g,
                                                 const float* __restrict__ ln_b,
                                                 float* __restrict__ yg) {
  __shared__ float red[256];
  __shared__ float mu_s, var_s;
  const int d = threadIdx.x;
  const int bl = blockIdx.x;              // b*4096 + l
  const int b = bl >> 12, l = bl & 4095;
  const int q = ((l & 63) << 6) | (l >> 6);   // (h,w)<->(w,h) involution
  const size_t base = (size_t)b * KDIR * D_INN * LLx;
  const size_t dl = (size_t)d * LLx;
  float y = out_y[base + dl + l]
          + out_y[base + (size_t)1 * D_INN * LLx + dl + q]
          + out_y[base + (size_t)2 * D_INN * LLx + dl + (LLx - 1 - l)]
          + out_y[base + (size_t)3 * D_INN * LLx + dl + (LLx - 1 - q)];
  // mean over d=192
  red[d] = y;
  if (d < 64) red[192 + d] = 0.f;
  __syncthreads();
  for (int s = 128; s > 0; s >>= 1) {
    if (d < s) red[d] += red[d + s];
    __syncthreads();
  }
  if (d == 0) mu_s = red[0] * (1.f / 192.f);
  __syncthreads();
  const float c0 = y - mu_s;
  red[d] = c0 * c0;
  if (d < 64) red[192 + d] = 0.f;
  __syncthreads();
  for (int s = 128; s > 0; s >>= 1) {
    if (d < s) red[d] += red[d + s];
    __syncthreads();
  }
  if (d == 0) var_s = red[0] * (1.f / 192.f);
  __syncthreads();
  const float yn = c0 * rsqrtf(var_s + 1e-5f) * ln_g[d] + ln_b[d];
  const float zv = z[(size_t)bl * D_INN + d];
  yg[(size_t)bl * D_INN + d] = yn * silu_f(zv);
}

// ---------------------------------------------------------------------------
// Small prep kernels: global-average pool, fused weight M2, per-(b,o) bias
// ---------------------------------------------------------------------------
__global__ __launch_bounds__(256) void k_pool(const float* __restrict__ x,
                                              float* __restrict__ pool) {
  __shared__ float red[256];
  const int bc = blockIdx.x;              // b*96 + c
  const int t = threadIdx.x;
  const float* src = x + (size_t)bc * LLx;
  float s = 0.f;
  for (int i = t; i < LLx; i += 256) s += src[i];
  red[t] = s;
  __syncthreads();
  for (int sh = 128; sh > 0; sh >>= 1) {
    if (t < sh) red[t] += red[t + sh];
    __syncthreads();
  }
  if (t == 0) pool[bc] = red[0] * (1.f / (float)LLx);
}

__global__ void k_m2(const float* __restrict__ proj_w, const float* __restrict__ W_out,
                     float* __restrict__ M2) {
  const int idx = blockIdx.x * blockDim.x + threadIdx.x;   // o*192 + d
  if (idx >= C_INx * D_INN) return;
  const int dd = idx % D_INN, o = idx / D_INN;
  float s = 0.f;
  for (int c = 0; c < C_INx; ++c)
    s += proj_w[o * (2 * C_INx) + c] * W_out[c * D_INN + dd];
  M2[idx] = s;
}

__global__ void k_bconst(const float* __restrict__ proj_w, const float* __restrict__ pool,
                         float* __restrict__ bconst) {
  const int idx = blockIdx.x * blockDim.x + threadIdx.x;   // b*96 + o
  if (idx >= NB * C_INx) return;
  const int o = idx % C_INx, b = idx / C_INx;
  float s = 0.f;
  for (int c = 0; c < C_INx; ++c)
    s += proj_w[o * (2 * C_INx) + C_INx + c] * pool[b * C_INx + c];
  bconst[idx] = s;
}

// ---------------------------------------------------------------------------
// Kernel 6: out = relu(BN(yg @ M2.T + bconst))   (M=16384, N=96, K=192)
// fp32 WMMA 16x16x4 with fused BN + ReLU epilogue.
// ---------------------------------------------------------------------------
__global__ __launch_bounds__(128) void k_gemm_out(const float* __restrict__ yg,
                                                  const float* __restrict__ M2,
                                                  const float* __restrict__ bconst,
                                                  const float* __restrict__ bn_g,
                                                  const float* __restrict__ bn_b,
                                                  const float* __restrict__ bn_mean,
                                                  const float* __restrict__ bn_var,
                                                  float* __restrict__ out) {
  const int lane = threadIdx.x & 31;
  const int tile = blockIdx.x * 4 + (threadIdx.x >> 5);
  const int NT = C_INx / 16;              // 6 column tiles
  const int mt = tile / NT, nt = tile % NT;
  const int m0 = mt * 16, n0 = nt * 16;
  const int half = lane >> 4, mr = lane & 15;
  const int m = m0 + mr;
  const int ncol = n0 + mr;
  v8f acc = {};
  for (int kb = 0; kb < D_INN; kb += 4) {
    const int k0 = kb + 2 * half;
    v2f a, bv;
    a.x  = yg[(size_t)m * D_INN + k0];
    a.y  = yg[(size_t)m * D_INN + k0 + 1];
    bv.x = M2[ncol * D_INN + k0];
    bv.y = M2[ncol * D_INN + k0 + 1];
    acc = __builtin_amdgcn_wmma_f32_16x16x4_f32(false, a, false, bv,
                                                (short)0, acc, false, false);
  }
  const int o = n0 + mr;
  const float sc = bn_g[o] * rsqrtf(bn_var[o] + 1e-5f);
  const float mn = bn_mean[o];
  const float bb = bn_b[o];
  for (int r = 0; r < 8; ++r) {
    const int mrow = m0 + r + 8 * half;
    const int b = mrow >> 12, l = mrow & 4095;
    float v = acc[r] + bconst[b * C_INx + o];
    v = (v - mn) * sc + bb;
    out[((size_t)b * C_INx + o) * LLx + l] = fmaxf(v, 0.f);
  }
}

// ---------------------------------------------------------------------------
extern "C" void kernel_launch(void* const* d_in, const int* in_sizes, int n_in,
                              void* d_out, int out_size, void* d_ws, size_t ws_size,
                              hipStream_t stream) {
  const float* x       = (const float*)d_in[0];
  const float* W_in    = (const float*)d_in[1];
  const float* conv_w  = (const float*)d_in[2];
  const float* conv_b  = (const float*)d_in[3];
  const float* xpw     = (const float*)d_in[4];
  const float* dt_w    = (const float*)d_in[5];
  const float* dt_b    = (const float*)d_in[6];
  const float* A_logs  = (const float*)d_in[7];
  const float* Ds      = (const float*)d_in[8];
  const float* ln_g    = (const float*)d_in[9];
  const float* ln_b    = (const float*)d_in[10];
  const float* W_out   = (const float*)d_in[11];
  const float* proj_w  = (const float*)d_in[12];
  const float* bn_g    = (const float*)d_in[13];
  const float* bn_b    = (const float*)d_in[14];
  const float* bn_mean = (const float*)d_in[15];
  const float* bn_var  = (const float*)d_in[16];
  float* out = (float*)d_out;

  // Workspace layout (floats). Total ~= 24.53M floats (~98 MB).
  float* ws     = (float*)d_ws;
  float* x_in   = ws;                                 // 4*192*4096 = 3,145,728
  float* z      = x_in + (size_t)NB * D_INN * LLx;    // 3,145,728
  float* xc     = z    + (size_t)NB * LLx * D_INN;    // 3,145,728
  float* xdbl   = xc   + (size_t)NB * D_INN * LLx;    // 16*38*4096 = 2,490,368
  float* out_y  = xdbl + (size_t)NB * KDIR * NROW * LLx;   // 16*192*4096 = 12,582,912
  float* pool   = out_y + (size_t)NB * KDIR * D_INN * LLx; // 384
  float* M2     = pool + NB * C_INx;                  // 18,432
  float* bconst = M2 + C_INx * D_INN;                 // 384
  float* yg     = x_in;   // alias: x_in is dead after k_conv, yg written after k_scan

  // 1) input projection (WMMA): 24576 tiles, 4 waves/block
  k_gemm_in<<<(1024 * 24) / 4, 128, 0, stream>>>(x, W_in, x_in, z);
  // 2) depthwise conv + SiLU
  {
    const int n = NB * D_INN * LLx;
    k_conv<<<(n + 255) / 256, 256, 0, stream>>>(x_in, conv_w, conv_b, xc);
  }
  // prep for output stage (independent of scan chain)
  k_pool<<<NB * C_INx, 256, 0, stream>>>(x, pool);
  k_m2<<<(C_INx * D_INN + 255) / 256, 256, 0, stream>>>(proj_w, W_out, M2);
  k_bconst<<<(NB * C_INx + 255) / 256, 256, 0, stream>>>(proj_w, pool, bconst);
  // 3) per-direction projections
  {
    const int n = NB * KDIR * NROW * LLx;
    k_xdbl<<<(n + 255) / 256, 256, 0, stream>>>(xc, xpw, xdbl);
  }
  // 4) selective scan: 16 blocks x 192 lanes = 3072 recurrences
  k_scan<<<NB * KDIR, 192, 0, stream>>>(xc, xdbl, dt_w, dt_b, A_logs, Ds, out_y);
  // 5) merge + LN + gate
  k_combine<<<NB * LLx, 192, 0, stream>>>(out_y, z, ln_g, ln_b, yg);
  // 6) fused output GEMM + BN + ReLU (WMMA): 6144 tiles, 4 waves/block
  k_gemm_out<<<(1024 * 6) / 4, 128, 0, stream>>>(yg, M2, bconst, bn_g, bn_b,
                                                 bn_mean, bn_var, out);
}